// AttackNet_26903675142297
// MI455X (gfx1250) — compile-verified
//
#include <hip/hip_runtime.h>
#include <hip/hip_bf16.h>

// Problem constants
#define NSAMP 32
#define CIN 3
#define HW 50176            // 224*224
#define OUTCH 3
#define PLANES (NSAMP * OUTCH)   // 96 (n, channel) planes

typedef __attribute__((ext_vector_type(2))) float v2f;
typedef __attribute__((ext_vector_type(8))) float v8f;

// Map float -> ordered unsigned so integer atomicMin/Max give float min/max.
__device__ __forceinline__ unsigned f2ord(float f) {
    unsigned u = __float_as_uint(f);
    return (u & 0x80000000u) ? ~u : (u | 0x80000000u);
}
__device__ __forceinline__ float ord2f(unsigned u) {
    return (u & 0x80000000u) ? __uint_as_float(u & 0x7FFFFFFFu)
                             : __uint_as_float(~u);
}

// ---------------------------------------------------------------------------
// Kernel 0: initialize min/max slots in workspace (must run every launch:
// harness does not re-poison between replays and we must be deterministic).
// ---------------------------------------------------------------------------
__global__ void attack_init_ws(unsigned* __restrict__ wmin,
                               unsigned* __restrict__ wmax) {
    int i = threadIdx.x;
    if (i < PLANES) {
        wmin[i] = 0xFFFFFFFFu;  // encodes +inf side
        wmax[i] = 0u;           // encodes -inf side
    }
}

// ---------------------------------------------------------------------------
// Kernel 1: target-class 1x1 conv via V_WMMA_F32_16X16X4_F32 (+bias),
// write unnormalized pert to d_out, reduce per-(n,ch) min/max.
// Grid: (49, 32), block: 256 threads (8 wave32 waves).
// 49 blocks * 8 waves = 392 waves; each wave does 8 chunks of 16 pixels
// -> 392*8*16 = 50176 pixels per sample, exact cover, no divergence.
// ---------------------------------------------------------------------------
__global__ void __launch_bounds__(256)
attack_conv_minmax(const float* __restrict__ x,
                   const int*   __restrict__ target,
                   const float* __restrict__ Wm,     // (60,3) row-major
                   const float* __restrict__ bias,   // (60,)
                   float*       __restrict__ out,
                   unsigned*    __restrict__ wmin,
                   unsigned*    __restrict__ wmax) {
    const int n      = blockIdx.y;
    const int tid    = threadIdx.x;
    const int lane   = tid & 31;
    const int waveId = tid >> 5;
    const bool hi    = (lane & 16) != 0;  // false: K0/K1 side, true: K2/K3 side
    const int col    = lane & 15;         // pixel column / A-matrix row id

    const int tgt = target[n];
    const float* Wrow = Wm + (size_t)tgt * (OUTCH * CIN);  // 3 rows x 3 cols

    // --- A matrix (16x4): rows 0..2 = W rows of target class, rest zero ---
    // lane<16 : row M=col, VGPR0=K0, VGPR1=K1
    // lane>=16: row M=col, VGPR0=K2, VGPR1=K3(=0)
    v2f a;
    float a0 = 0.0f, a1 = 0.0f;
    if (col < OUTCH) {
        if (!hi) { a0 = Wrow[col * CIN + 0]; a1 = Wrow[col * CIN + 1]; }
        else     { a0 = Wrow[col * CIN + 2]; a1 = 0.0f; }
    }
    a.x = a0; a.y = a1;

    const float bsc0 = bias[tgt * OUTCH + 0];
    const float bsc1 = bias[tgt * OUTCH + 1];
    const float bsc2 = bias[tgt * OUTCH + 2];

    const float* x0 = x + (size_t)n * CIN * HW;
    const float* x1 = x0 + HW;
    const float* x2 = x1 + HW;
    float* o0 = out + (size_t)n * OUTCH * HW;
    float* o1 = o0 + HW;
    float* o2 = o1 + HW;

    // Per-lane base pointer: one v_cndmask on the address, hoisted out of
    // the loop, so inner-loop loads are unconditional (no saveexec dance).
    const float* xa = hi ? x2 : x0;

    float lmin0 = __builtin_inff(), lmin1 = __builtin_inff(), lmin2 = __builtin_inff();
    float lmax0 = -__builtin_inff(), lmax1 = -__builtin_inff(), lmax2 = -__builtin_inff();

    const int globalWave = blockIdx.x * 8 + waveId;   // 0..391

    #pragma unroll
    for (int ck = 0; ck < 8; ++ck) {
        const int chunk = globalWave * 8 + ck;        // 0..3135
        const int pix   = chunk * 16 + col;           // 0..50175

        // --- B matrix (4x16): row K=c holds x[n,c,pixel 0..15] ---
        // Unconditional loads in all lanes; mask K3 row to zero via cndmask.
        const float vb0 = xa[pix];                    // K0 (lo) / K2 (hi)
        const float vb1 = x1[pix];                    // K1 row data
        v2f bm;
        bm.x = vb0;
        bm.y = hi ? 0.0f : vb1;                       // K3 row is zero
        v8f cacc = {};                                // zero accumulator

        // D = A x B + 0  (full f32 precision)
        v8f d = __builtin_amdgcn_wmma_f32_16x16x4_f32(
            /*neg_a=*/false, a, /*neg_b=*/false, bm,
            /*c_mod=*/(short)0, cacc,
            /*reuse_a=*/false, /*reuse_b=*/false);

        if (!hi) {
            // D VGPR m, lanes 0-15 -> channel m, pixel = chunk*16 + lane
            float p0 = d[0] + bsc0;
            float p1 = d[1] + bsc1;
            float p2 = d[2] + bsc2;
            o0[pix] = p0; o1[pix] = p1; o2[pix] = p2;
            lmin0 = fminf(lmin0, p0); lmax0 = fmaxf(lmax0, p0);
            lmin1 = fminf(lmin1, p1); lmax1 = fmaxf(lmax1, p1);
            lmin2 = fminf(lmin2, p2); lmax2 = fmaxf(lmax2, p2);
        }
    }

    // --- block-level reduction (LDS), then 3x2 global atomics per block ---
    __shared__ float smin[OUTCH][256];
    __shared__ float smax[OUTCH][256];
    smin[0][tid] = lmin0; smin[1][tid] = lmin1; smin[2][tid] = lmin2;
    smax[0][tid] = lmax0; smax[1][tid] = lmax1; smax[2][tid] = lmax2;
    __syncthreads();
    #pragma unroll
    for (int s = 128; s > 0; s >>= 1) {
        if (tid < s) {
            #pragma unroll
            for (int c = 0; c < OUTCH; ++c) {
                smin[c][tid] = fminf(smin[c][tid], smin[c][tid + s]);
                smax[c][tid] = fmaxf(smax[c][tid], smax[c][tid + s]);
            }
        }
        __syncthreads();
    }
    if (tid < OUTCH) {
        atomicMin(&wmin[n * OUTCH + tid], f2ord(smin[tid][0]));
        atomicMax(&wmax[n * OUTCH + tid], f2ord(smax[tid][0]));
    }
}

// ---------------------------------------------------------------------------
// Kernel 2: in-place normalize: 2*(p - pmin)/(pmax - pmin) - 1   (scale = 1.0)
// float4 vectorized; HW % 4 == 0 so a float4 never crosses a plane boundary.
// ---------------------------------------------------------------------------
__global__ void __launch_bounds__(256)
attack_normalize(float* __restrict__ out,
                 const unsigned* __restrict__ wmin,
                 const unsigned* __restrict__ wmax) {
    const int total4 = PLANES * (HW / 4);
    int idx = blockIdx.x * blockDim.x + threadIdx.x;
    if (idx >= total4) return;
    const int plane = (idx * 4) / HW;          // (n, ch) plane id
    const float pmin = ord2f(wmin[plane]);
    const float pmax = ord2f(wmax[plane]);
    const float inv  = 2.0f / (pmax - pmin);

    float4 v = reinterpret_cast<float4*>(out)[idx];
    v.x = (v.x - pmin) * inv - 1.0f;
    v.y = (v.y - pmin) * inv - 1.0f;
    v.z = (v.z - pmin) * inv - 1.0f;
    v.w = (v.w - pmin) * inv - 1.0f;
    reinterpret_cast<float4*>(out)[idx] = v;
}

// ---------------------------------------------------------------------------
extern "C" void kernel_launch(void* const* d_in, const int* in_sizes, int n_in,
                              void* d_out, int out_size, void* d_ws, size_t ws_size,
                              hipStream_t stream) {
    const float* x      = (const float*)d_in[0];   // (32,3,224,224)
    const int*   target = (const int*)  d_in[1];   // (32,)
    const float* Wm     = (const float*)d_in[2];   // (60,3,1,1)
    const float* bias   = (const float*)d_in[3];   // (60,)
    float* out = (float*)d_out;                    // (32,3,224,224)

    unsigned* wmin = (unsigned*)d_ws;              // 96 slots
    unsigned* wmax = wmin + PLANES;                // 96 slots

    attack_init_ws<<<1, 128, 0, stream>>>(wmin, wmax);

    attack_conv_minmax<<<dim3(49, NSAMP), 256, 0, stream>>>(
        x, target, Wm, bias, out, wmin, wmax);

    const int total4 = PLANES * (HW / 4);
    attack_normalize<<<(total4 + 255) / 256, 256, 0, stream>>>(out, wmin, wmax);
}